// ChamferLoss_6528350290147
// MI455X (gfx1250) — compile-verified
//
#include <hip/hip_runtime.h>

typedef __attribute__((ext_vector_type(2))) float v2f;
typedef __attribute__((ext_vector_type(8))) float v8f;

#define NPTS        8192
#define NB          4
#define CHUNK_PTS   2048                    // database points staged per chunk
#define CHUNK_TILES (CHUNK_PTS / 16)        // 128 WMMA tiles per chunk
#define NCHUNK      (NPTS / CHUNK_PTS)      // 4 chunks
#define TOTAL_MINS  (2 * NB * NPTS)         // 65536 per-point mins

// v_min3_num_f32 directly: avoids the per-element canonicalize (v_max_num v,v,v)
// that fminf chains emit.
__device__ __forceinline__ float min3f(float a, float b, float c) {
    float r;
    asm("v_min3_num_f32 %0, %1, %2, %3" : "=v"(r) : "v"(a), "v"(b), "v"(c));
    return r;
}

// One wave handles 16 query points; all 8 waves of a block share one batch and
// sweep the same database, which is staged once per block into LDS in the
// exact A-matrix VGPR layout of V_WMMA_F32_16X16X4_F32:
//   lane m (0-15):  (bx, by)   = K0,K1
//   lane m+16:      (bz, |b|^2)= K2,K3
// B column n = (-2qx, -2qy, -2qz, 1); |q|^2 added after the min.
__global__ __launch_bounds__(256) void chamfer_min_kernel(
    const float* __restrict__ q_pts,   // [NB, NPTS, 3] queries
    const float* __restrict__ db_pts,  // [NB, NPTS, 3] database
    float* __restrict__ out_min)       // [NB * NPTS] min sq-dist per query
{
    __shared__ v2f lds2[CHUNK_TILES * 32];  // 32 KB pre-swizzled A tiles

    const int lane  = threadIdx.x & 31;
    const int wave  = threadIdx.x >> 5;
    const int batch = blockIdx.x >> 6;               // 64 blocks per batch
    const int qg    = ((blockIdx.x & 63) << 3) + wave;
    const int m16   = lane & 15;
    const bool hi   = lane >= 16;

    const float* qb = q_pts  + (size_t)batch * NPTS * 3 + (size_t)qg * 48;
    const float* db = db_pts + (size_t)batch * NPTS * 3;

    // This lane's query column (column index = lane & 15).
    const float qx = qb[3 * m16 + 0];
    const float qy = qb[3 * m16 + 1];
    const float qz = qb[3 * m16 + 2];
    const float q2 = qx * qx + qy * qy + qz * qz;

    v2f bmat;
    bmat.x = hi ? (-2.0f * qz) : (-2.0f * qx);
    bmat.y = hi ? 1.0f         : (-2.0f * qy);

    v8f runmin;
#pragma unroll
    for (int i = 0; i < 8; ++i) runmin[i] = 3.4e38f;

    for (int c = 0; c < NCHUNK; ++c) {
        // ---- Stage chunk c: swizzle into A layout, precompute |b|^2 once ----
        const float* cb = db + (size_t)c * CHUNK_PTS * 3;
#pragma unroll
        for (int it = 0; it < CHUNK_PTS / 256; ++it) {
            const int   j  = it * 256 + threadIdx.x;
            const float x  = cb[3 * j + 0];
            const float y  = cb[3 * j + 1];
            const float z  = cb[3 * j + 2];
            const float n2 = x * x + y * y + z * z;
            const int   tl = j >> 4;
            const int   m  = j & 15;
            lds2[tl * 32 + m]      = (v2f){x, y};
            lds2[tl * 32 + m + 16] = (v2f){z, n2};
        }
        __syncthreads();

        // Overlap: prefetch next chunk's raw points while we crunch this one.
        if (c + 1 < NCHUNK) {
            const float* np = db + (size_t)(c + 1) * CHUNK_PTS * 3;
            __builtin_prefetch(np + threadIdx.x * 6, 0, 0);             // 6 KB
            __builtin_prefetch(np + 1536 + threadIdx.x * 6, 0, 0);      // 6 KB
            __builtin_prefetch(np + 3072 + threadIdx.x * 6, 0, 0);
            __builtin_prefetch(np + 4608 + threadIdx.x * 6, 0, 0);
        }

        // ---- Main sweep: ds_load_b64 + WMMA + min3, unrolled by 2 ----
        for (int t = 0; t < CHUNK_TILES; t += 2) {
            const v2f a0 = lds2[(t + 0) * 32 + lane];
            const v2f a1 = lds2[(t + 1) * 32 + lane];
            v8f cz = {};
            v8f d0 = __builtin_amdgcn_wmma_f32_16x16x4_f32(
                false, a0, false, bmat, (short)0, cz, false, false);
            v8f d1 = __builtin_amdgcn_wmma_f32_16x16x4_f32(
                false, a1, false, bmat, (short)0, cz, false, false);
#pragma unroll
            for (int i = 0; i < 8; ++i)
                runmin[i] = min3f(runmin[i], d0[i], d1[i]);
        }
        __syncthreads();  // protect LDS before next chunk restages
    }

    // Reduce 8 in-lane row elements, then combine the two row halves
    // (lane l <-> lane l^16 hold the same query column).
    float m = min3f(min3f(runmin[0], runmin[1], runmin[2]),
                    min3f(runmin[3], runmin[4], runmin[5]),
                    fminf(runmin[6], runmin[7]));
    m = fminf(m, __shfl_xor(m, 16, 32));

    const float d2 = fmaxf(m + q2, 0.0f);
    if (!hi) out_min[(size_t)batch * NPTS + (size_t)qg * 16 + m16] = d2;
}

// Deterministic single-block reduction: fixed-order strided sums + LDS tree.
__global__ __launch_bounds__(256) void chamfer_reduce_kernel(
    const float* __restrict__ mins, float* __restrict__ out)
{
    __shared__ float sm[256];
    const int tid = threadIdx.x;
    float s = 0.0f;
    for (int i = tid; i < TOTAL_MINS; i += 256) s += mins[i];
    sm[tid] = s;
    __syncthreads();
    for (int off = 128; off > 0; off >>= 1) {
        if (tid < off) sm[tid] += sm[tid + off];
        __syncthreads();
    }
    if (tid == 0) out[0] = sm[0] / (float)(NB * NPTS);  // mean_x + mean_y
}

extern "C" void kernel_launch(void* const* d_in, const int* in_sizes, int n_in,
                              void* d_out, int out_size, void* d_ws, size_t ws_size,
                              hipStream_t stream) {
    const float* pred   = (const float*)d_in[0];
    const float* target = (const float*)d_in[1];
    float* mins = (float*)d_ws;              // 65536 floats = 256 KB scratch
    float* out  = (float*)d_out;

    // Pass 1: pred -> target mins (cham_x); 256 blocks x 8 waves = 2048 groups
    chamfer_min_kernel<<<256, 256, 0, stream>>>(pred, target, mins);
    // Pass 2: target -> pred mins (cham_y)
    chamfer_min_kernel<<<256, 256, 0, stream>>>(target, pred, mins + NB * NPTS);
    // Final deterministic reduction to the scalar loss.
    chamfer_reduce_kernel<<<1, 256, 0, stream>>>(mins, out);
}